// NCEAverage_4569845203285
// MI455X (gfx1250) — compile-verified
//
#include <hip/hip_runtime.h>
#include <hip/hip_bf16.h>
#include <math.h>

// Problem constants from the reference
#define BATCH   64
#define DIMS    128
#define NROWS   500000
#define KP1     4096
#define SCORES  ((size_t)BATCH * KP1)        // 262144
#define BANKELEMS ((size_t)NROWS * DIMS)     // 64,000,000

typedef __attribute__((ext_vector_type(2))) float v2f;
typedef __attribute__((ext_vector_type(8))) float v8f;

// ---------------------------------------------------------------------------
// Score kernel: one wave32 computes 16 candidate scores for one batch b using
// V_WMMA_F32_16X16X4_F32 accumulated over D=128 (32 steps of K=4).
//
// A (16x4 f32): lane l<16 holds row M=l at K = {0,1}; lane l>=16 holds row
// M=l-16 at K = {2,3}  -> per-lane float2 at column offset 2*(l>>4).
// B (4x16 f32): x[b] chunk replicated across all N columns (layout mirrors A's
// K mapping), so every C column carries the same dot products.
// C (16x16 f32): lane 0 VGPRs 0..7 = M 0..7 (N=0); lane 16 = M 8..15.
// ---------------------------------------------------------------------------
__global__ __launch_bounds__(256) void nce_score_kernel(
    const float* __restrict__ mem_gene, const float* __restrict__ mem_image,
    const float* __restrict__ image,    const float* __restrict__ gene,
    const int*   __restrict__ idx,      float* __restrict__ out_scores) {
  const int wave = threadIdx.x >> 5;
  const int lane = threadIdx.x & 31;
  const int tile = blockIdx.x * 8 + wave;      // 0 .. 16383
  const int b     = tile >> 8;                 // batch
  const int kbase = (tile & 255) << 4;         // candidate tile base (16 wide)

  // out_image = softmax(mem_gene[idx] . image); out_gene = softmax(mem_image[idx] . gene)
  const float* W = (blockIdx.y == 0) ? mem_gene : mem_image;
  const float* X = (blockIdx.y == 0) ? image    : gene;
  float* O = out_scores + (size_t)blockIdx.y * SCORES;

  const int m    = lane & 15;                  // candidate row within tile
  const int coff = (lane >> 4) << 1;           // 0 for lanes 0-15, 2 for 16-31

  const int row = idx[b * KP1 + kbase + m];    // gathered bank row
  const float* Wr = W + (size_t)row * DIMS + coff;
  const float* Xr = X + (size_t)b   * DIMS + coff;

  v8f acc = {};
#pragma unroll 4
  for (int d0 = 0; d0 < DIMS; d0 += 4) {
    v2f a  = *(const v2f*)(Wr + d0);
    v2f bb = *(const v2f*)(Xr + d0);
    // (neg_a, A, neg_b, B, c_mod, C, reuse_a, reuse_b)
    acc = __builtin_amdgcn_wmma_f32_16x16x4_f32(
        false, a, false, bb, (short)0, acc, false, false);
  }

  if (lane == 0) {
#pragma unroll
    for (int v = 0; v < 8; ++v) O[(size_t)b * KP1 + kbase + v] = acc[v];
  } else if (lane == 16) {
#pragma unroll
    for (int v = 0; v < 8; ++v) O[(size_t)b * KP1 + kbase + 8 + v] = acc[v];
  }
}

// ---------------------------------------------------------------------------
// Stabilized softmax over each row of 4096 scores (in place in d_out).
// One 256-thread block per row; 128 rows cover both outputs.
// ---------------------------------------------------------------------------
__global__ __launch_bounds__(256) void nce_softmax_kernel(float* __restrict__ data) {
  float* row = data + (size_t)blockIdx.x * KP1;
  __shared__ float red[256];
  const int t = threadIdx.x;

  float mx = -INFINITY;
  for (int i = t; i < KP1; i += 256) mx = fmaxf(mx, row[i]);
  red[t] = mx;
  __syncthreads();
  for (int s = 128; s > 0; s >>= 1) {
    if (t < s) red[t] = fmaxf(red[t], red[t + s]);
    __syncthreads();
  }
  mx = red[0];
  __syncthreads();

  float sum = 0.0f;
  for (int i = t; i < KP1; i += 256) sum += expf(row[i] - mx);
  red[t] = sum;
  __syncthreads();
  for (int s = 128; s > 0; s >>= 1) {
    if (t < s) red[t] += red[t + s];
    __syncthreads();
  }
  const float inv = 1.0f / red[0];

  for (int i = t; i < KP1; i += 256) row[i] = expf(row[i] - mx) * inv;
}

// ---------------------------------------------------------------------------
// Momentum update of the 64 touched bank rows (banks already bulk-copied into
// d_out by hipMemcpyAsync): pos = 0.5*mem_old + 0.5*x; pos /= ||pos||2.
// One 128-thread block per (b, bank).
// ---------------------------------------------------------------------------
__global__ __launch_bounds__(128) void nce_momentum_kernel(
    const float* __restrict__ mem_image, const float* __restrict__ mem_gene,
    const float* __restrict__ image,     const float* __restrict__ gene,
    const int*   __restrict__ index,
    float* __restrict__ out_mem_image,   float* __restrict__ out_mem_gene) {
  const float* mem = (blockIdx.y == 0) ? mem_image : mem_gene;
  const float* x   = (blockIdx.y == 0) ? image     : gene;
  float*       o   = (blockIdx.y == 0) ? out_mem_image : out_mem_gene;

  const int b = blockIdx.x;
  const int d = threadIdx.x;          // 0..127
  const int r = index[b];

  const float v = 0.5f * mem[(size_t)r * DIMS + d] + 0.5f * x[(size_t)b * DIMS + d];

  __shared__ float red[128];
  red[d] = v * v;
  __syncthreads();
  for (int s = 64; s > 0; s >>= 1) {
    if (d < s) red[d] += red[d + s];
    __syncthreads();
  }
  o[(size_t)r * DIMS + d] = v / sqrtf(red[0]);
}

// ---------------------------------------------------------------------------
// Launch: scores (WMMA) -> softmax -> bank copies (async d2d) -> row updates.
// d_out layout: [out_image 64x4096][out_gene 64x4096][new_mem_image][new_mem_gene]
// ---------------------------------------------------------------------------
extern "C" void kernel_launch(void* const* d_in, const int* in_sizes, int n_in,
                              void* d_out, int out_size, void* d_ws, size_t ws_size,
                              hipStream_t stream) {
  const float* image     = (const float*)d_in[0];
  const float* gene      = (const float*)d_in[1];
  const float* mem_image = (const float*)d_in[2];
  const float* mem_gene  = (const float*)d_in[3];
  const int*   index     = (const int*)d_in[4];
  const int*   idx       = (const int*)d_in[5];

  float* out          = (float*)d_out;
  float* out_scores   = out;                       // 2 * 262144
  float* out_memimage = out + 2 * SCORES;          // 64,000,000
  float* out_memgene  = out + 2 * SCORES + BANKELEMS;

  // Raw scores via f32 WMMA: 2048 blocks x 8 waves = 16384 (b, ktile) waves per
  // output; grid.y selects which of the two score matrices.
  dim3 gscore(2048, 2, 1);
  nce_score_kernel<<<gscore, 256, 0, stream>>>(mem_gene, mem_image, image, gene,
                                               idx, out_scores);

  // Softmax across the 4096 candidates of each of the 128 rows.
  nce_softmax_kernel<<<128, 256, 0, stream>>>(out_scores);

  // Bulk-copy the banks into the output (bandwidth floor), then patch 64 rows.
  hipMemcpyAsync(out_memimage, mem_image, BANKELEMS * sizeof(float),
                 hipMemcpyDeviceToDevice, stream);
  hipMemcpyAsync(out_memgene, mem_gene, BANKELEMS * sizeof(float),
                 hipMemcpyDeviceToDevice, stream);

  dim3 gmom(BATCH, 2, 1);
  nce_momentum_kernel<<<gmom, 128, 0, stream>>>(mem_image, mem_gene, image, gene,
                                                index, out_memimage, out_memgene);
}